// TransducerJoint_66915590472509
// MI455X (gfx1250) — compile-verified
//
#include <hip/hip_runtime.h>

// Problem constants (from the reference): B=16, T=256, U=96, H=512, fp32.
#define BB 16
#define TT 256
#define UU 96
#define HH 512

typedef float vf4 __attribute__((ext_vector_type(4)));

// One workgroup (256 threads, 8 wave32s) per (b,t) row of the output.
// Store-bandwidth-bound: 805 MB written per call -> floor ~35us @ 23.3 TB/s.
// f[b,t,:] staged via CDNA5 async global->LDS DMA; g[b] is L2-resident.
__global__ __launch_bounds__(256) void transducer_joint_kernel(
    const float* __restrict__ f,      // (B,T,H)
    const float* __restrict__ g,      // (B,U,H)
    const int*   __restrict__ f_len,  // (B,)
    const int*   __restrict__ g_len,  // (B,)
    float*       __restrict__ out)    // (B,T,U,H)
{
    __shared__ float fsh[HH];

    const int tid = threadIdx.x;
    const int bt  = blockIdx.x;        // b*T + t
    const int b   = bt / TT;
    const int t   = bt - b * TT;

    // ---- Stage f[b,t,0:512] (2048 B) into LDS with the CDNA5 async
    // global->LDS path (ASYNCcnt). 128 lanes x 16 B; waves 0..3 are fully
    // active inside the branch, so the async op never runs with partial EXEC.
    if (tid < (HH * 4) / 16) {
        unsigned lds_off = (unsigned)(unsigned long long)(const void*)fsh
                         + (unsigned)(tid * 16);
        const float* gp = f + (size_t)bt * HH + tid * 4;
        asm volatile("global_load_async_to_lds_b128 %0, %1, off"
                     :: "v"(lds_off), "v"(gp)
                     : "memory");
    }
    asm volatile("s_wait_asynccnt 0" ::: "memory");
    __syncthreads();

    // Valid region is a prefix of the flattened (u,h) space:
    //   u < g_len[b]  <=>  base < g_len[b]*H.   Block-uniform scalar.
    const int validN = (t < f_len[b]) ? (g_len[b] << 9) : 0;

    const float* __restrict__ gb = g   + (size_t)b  * (UU * HH);
    float*       __restrict__ ob = out + (size_t)bt * (UU * HH);

    // base = tid*4 + k*1024; (base & 511) is loop-invariant -> load this
    // thread's f-fragment from LDS exactly once.
    const vf4 fv = *reinterpret_cast<const vf4*>(&fsh[(tid * 4) & (HH - 1)]);

    // U*H = 49152 floats; 256 threads x float4 = 1024 floats/iter -> 48 iters.
    // g[b] shares the (u,h) flattening, so its address is simply gb + base.
    #pragma unroll 8
    for (int base = tid * 4; base < UU * HH; base += 256 * 4) {
        vf4 v = (vf4)0.0f;
        if (base < validN) {
            v = fv + *reinterpret_cast<const vf4*>(gb + base);
        }
        // Write-only 805 MB stream: non-temporal 128-bit stores.
        __builtin_nontemporal_store(v, reinterpret_cast<vf4*>(ob + base));
    }
}

extern "C" void kernel_launch(void* const* d_in, const int* in_sizes, int n_in,
                              void* d_out, int out_size, void* d_ws, size_t ws_size,
                              hipStream_t stream) {
    const float* f     = (const float*)d_in[0];
    const float* g     = (const float*)d_in[1];
    const int*   f_len = (const int*)d_in[2];
    const int*   g_len = (const int*)d_in[3];
    float*       out   = (float*)d_out;

    dim3 grid(BB * TT);   // 4096 blocks, one per (b,t)
    dim3 block(256);      // 8 wave32s
    hipLaunchKernelGGL(transducer_joint_kernel, grid, block, 0, stream,
                       f, g, f_len, g_len, out);
}